// CASMModelWrapper_27187142983781
// MI455X (gfx1250) — compile-verified
//
#include <hip/hip_runtime.h>
#include <math.h>

// Sizes fixed by the reference
#define B_ 8
#define S_ 2048
#define H_ 4096
#define NSLOTS 64
#define MEM_ 16
#define RH_ 256
#define TOPK 4

typedef float v2f  __attribute__((ext_vector_type(2)));
typedef float v8f  __attribute__((ext_vector_type(8)));
typedef float fv4  __attribute__((ext_vector_type(4)));

// ---------------------------------------------------------------------------
// Kernel 0: zero the padded query buffer [16, H] (rows 8..15 stay zero so the
// WMMA M=16 tile is valid).
// ---------------------------------------------------------------------------
__global__ void init_query_kernel(float* __restrict__ query) {
    int i = blockIdx.x * blockDim.x + threadIdx.x;   // 16*4096 = 65536 threads
    query[i] = 0.0f;
}

// ---------------------------------------------------------------------------
// Kernel 1: query[b,h] = mean_s embeds[b,s,h].  256 MB streamed once -> use
// non-temporal loads. Grid (8 b, 16 h-chunks, 8 s-chunks) = 1024 blocks so the
// whole chip participates; partial sums combined with f32 global atomics.
// ---------------------------------------------------------------------------
__global__ void mean_kernel(const float* __restrict__ embeds,
                            float* __restrict__ query) {
    const int b  = blockIdx.x;             // 0..7
    const int hc = blockIdx.y;             // 0..15
    const int sc = blockIdx.z;             // 0..7
    const int h  = hc * 256 + threadIdx.x; // 0..4095, coalesced

    const float* p = embeds + ((long)b * S_ + (long)sc * 256) * H_ + h;
    float acc = 0.0f;
    #pragma unroll 8
    for (int s = 0; s < 256; ++s) {
        acc += __builtin_nontemporal_load(p);   // streaming, don't pollute L2
        p += H_;
    }
    atomicAdd(&query[b * H_ + h], acc * (1.0f / (float)S_));
}

// ---------------------------------------------------------------------------
// Kernel 2: router. One workgroup of 512 threads (16 waves).
//  Stage 1: h = relu(query @ W1^T + b1) via V_WMMA_F32_16X16X4_F32.
//           Wave w owns RH columns [16w,16w+16); K-loop over H in steps of 4.
//  Stage 2: logits = h @ W2^T + b2  (512 dot products of length 256, scalar).
//  Stage 3: per-batch top-4 + softmax -> weights/ids to global scratch.
// ---------------------------------------------------------------------------
__global__ void router_kernel(const float* __restrict__ query,  // [16,H] padded
                              const float* __restrict__ W1,     // [RH,H]
                              const float* __restrict__ b1,     // [RH]
                              const float* __restrict__ W2,     // [N,RH]
                              const float* __restrict__ b2,     // [N]
                              float* __restrict__ weights_out,  // [B,K]
                              int*   __restrict__ ids_out) {    // [B,K]
    __shared__ float h_lds[16][RH_];        // 16 KB
    __shared__ float logits[B_][NSLOTS];    // 2 KB

    const int tid  = threadIdx.x;
    const int wave = tid >> 5;              // 0..15
    const int lane = tid & 31;
    const int half = lane >> 4;             // 0 | 1
    const int l16  = lane & 15;

    // ---- Stage 1: WMMA f32 16x16x4 ----
    // A layout (16x4 f32): lanes 0-15 -> M=lane, K={k0,k0+1}; lanes 16-31 ->
    // M=lane-16, K={k0+2,k0+3}.  B layout (4x16): lanes hold N=l16, same K split.
    {
        const int n = wave * 16 + l16;                 // RH column
        const float* aRow = query + (long)l16 * H_;    // M = l16
        const float* bRow = W1    + (long)n   * H_;    // B(k,n) = W1[n,k]
        v8f acc = {};
        for (int k0 = 0; k0 < H_; k0 += 4) {
            const int ko = k0 + half * 2;
            v2f a, b;
            a[0] = aRow[ko + 0];
            a[1] = aRow[ko + 1];
            b[0] = bRow[ko + 0];
            b[1] = bRow[ko + 1];
            acc = __builtin_amdgcn_wmma_f32_16x16x4_f32(
                false, a, false, b, (short)0, acc, false, false);
        }
        const float bias = b1[n];
        #pragma unroll
        for (int r = 0; r < 8; ++r) {
            const int m = r + half * 8;                // C/D: VGPR r -> M=r / r+8
            float v = acc[r] + bias;
            h_lds[m][n] = v > 0.0f ? v : 0.0f;         // ReLU
        }
    }
    __syncthreads();

    // ---- Stage 2: logits[b][n] (512 threads == 512 dot products) ----
    {
        const int b = tid >> 6;     // 0..7
        const int n = tid & 63;     // 0..63
        float acc = b2[n];
        const float* w2 = W2 + n * RH_;
        #pragma unroll 8
        for (int r = 0; r < RH_; ++r) acc += h_lds[b][r] * w2[r];
        logits[b][n] = acc;         // TEMP == 1.0
    }
    __syncthreads();

    // ---- Stage 3: top-4 + softmax, one thread per batch row ----
    if (tid < B_) {
        unsigned long long used = 0ull;
        float vals[TOPK]; int ids[TOPK];
        for (int k = 0; k < TOPK; ++k) {
            float best = -INFINITY; int bi = 0;
            for (int n = 0; n < NSLOTS; ++n) {
                if (!((used >> n) & 1ull) && logits[tid][n] > best) {
                    best = logits[tid][n]; bi = n;
                }
            }
            used |= (1ull << bi);
            vals[k] = best; ids[k] = bi;
        }
        float mx = vals[0];
        for (int k = 1; k < TOPK; ++k) mx = fmaxf(mx, vals[k]);
        float e[TOPK], s = 0.0f;
        for (int k = 0; k < TOPK; ++k) { e[k] = expf(vals[k] - mx); s += e[k]; }
        const float inv = 1.0f / s;
        for (int k = 0; k < TOPK; ++k) {
            weights_out[tid * TOPK + k] = e[k] * inv;
            ids_out[tid * TOPK + k]     = ids[k];
        }
    }
}

// ---------------------------------------------------------------------------
// Kernel 3: contrib[b,h] = sum_k w[b,k] * sum_m sigmoid(gate[slot,m]) *
// memory[slot,m,h].  Only selected slots are touched (<=8 MB vs reference's
// 16 MB all-slot pass). Grid (8, 16 h-chunks) x 256 threads, h coalesced.
// ---------------------------------------------------------------------------
__global__ void contrib_kernel(const float* __restrict__ gate_logits, // [N,M]
                               const float* __restrict__ memory,      // [N,M,H]
                               const float* __restrict__ weights,     // [B,K]
                               const int*   __restrict__ ids,         // [B,K]
                               float* __restrict__ contrib) {         // [B,H]
    const int b = blockIdx.x;
    const int h = blockIdx.y * 256 + threadIdx.x;
    float acc = 0.0f;
    #pragma unroll
    for (int k = 0; k < TOPK; ++k) {
        const int   slot = ids[b * TOPK + k];
        const float wk   = weights[b * TOPK + k];
        const float* gp  = gate_logits + slot * MEM_;
        const float* mp  = memory + ((long)slot * MEM_) * H_ + h;
        float sacc = 0.0f;
        #pragma unroll
        for (int m = 0; m < MEM_; ++m) {
            const float g = 1.0f / (1.0f + expf(-gp[m]));  // sigmoid
            sacc += g * mp[(long)m * H_];
        }
        acc += wk * sacc;
    }
    contrib[b * H_ + h] = acc;
}

// ---------------------------------------------------------------------------
// Kernel 4: out[b,s,h] = hidden[b,s,h] + contrib[b,h].  512 MB of streaming
// traffic -> float4 (B128) non-temporal loads/stores; contrib (128 KB) stays
// L2/WGP$ resident via regular loads.
// ---------------------------------------------------------------------------
__global__ void inject_kernel(const fv4* __restrict__ hidden,
                              const float* __restrict__ contrib,
                              fv4* __restrict__ out) {
    const long idx = (long)blockIdx.x * 256 + threadIdx.x;  // float4 index
    const long e   = idx << 2;                              // float index
    const int  h   = (int)(e & (H_ - 1));                   // H = 4096 = 2^12
    const int  b   = (int)(e >> 23);                        // S*H = 2^23
    fv4 hd = __builtin_nontemporal_load(&hidden[idx]);
    const fv4 c = *reinterpret_cast<const fv4*>(contrib + b * H_ + h);
    __builtin_nontemporal_store(hd + c, &out[idx]);
}

// ---------------------------------------------------------------------------
extern "C" void kernel_launch(void* const* d_in, const int* in_sizes, int n_in,
                              void* d_out, int out_size, void* d_ws, size_t ws_size,
                              hipStream_t stream) {
    const float* embeds      = (const float*)d_in[0];
    const float* hidden      = (const float*)d_in[1];
    const float* W1          = (const float*)d_in[2];
    const float* b1          = (const float*)d_in[3];
    const float* W2          = (const float*)d_in[4];
    const float* b2          = (const float*)d_in[5];
    const float* gate_logits = (const float*)d_in[6];
    const float* memory      = (const float*)d_in[7];
    // d_in[8] = top_k, fixed at 4

    // Workspace layout (bytes): query[16*H] | contrib[B*H] | weights[B*K] | ids[B*K]
    char* ws = (char*)d_ws;
    float* query   = (float*)(ws);                         // 262144 B
    float* contrib = (float*)(ws + 16 * H_ * 4);           // 131072 B
    float* weights = (float*)(ws + 16 * H_ * 4 + B_ * H_ * 4);
    int*   ids     = (int*)  (ws + 16 * H_ * 4 + B_ * H_ * 4 + B_ * TOPK * 4);

    // 0) zero padded query
    init_query_kernel<<<dim3(16 * H_ / 256), dim3(256), 0, stream>>>(query);

    // 1) mean over S (1024 blocks, NT streaming + atomic combine)
    mean_kernel<<<dim3(B_, H_ / 256, S_ / 256), dim3(256), 0, stream>>>(embeds, query);

    // 2) router: WMMA GEMM1 -> GEMM2 -> top-4 softmax (1 block, 16 waves)
    router_kernel<<<dim3(1), dim3(512), 0, stream>>>(query, W1, b1, W2, b2,
                                                     weights, ids);

    // 3) selected-slot contributions
    contrib_kernel<<<dim3(B_, H_ / 256), dim3(256), 0, stream>>>(
        gate_logits, memory, weights, ids, contrib);

    // 4) broadcast-add into hidden (64M floats as 16M float4)
    const long n4 = (long)B_ * S_ * H_ / 4;
    inject_kernel<<<dim3((unsigned)(n4 / 256)), dim3(256), 0, stream>>>(
        (const fv4*)hidden, contrib, (fv4*)d_out);
}